// AdjustInstanceArea_6339371728969
// MI455X (gfx1250) — compile-verified
//
#include <hip/hip_runtime.h>
#include <hip/hip_bf16.h>

// ---------------- problem constants (match reference) ----------------
constexpr int   NN_NODES   = 2000000;
constexpr int   NN_MOVABLE = 1500000;
constexpr int   NN_FILLER  = 400000;
constexpr int   NN_NETS    = 1500000;
constexpr int   NN_PINS    = NN_NETS * 4;
constexpr int   NBX        = 512;
constexpr float BINW       = 1000.0f / 512.0f;   // 1.953125 exact
constexpr float FXS        = 1048576.0f;         // 2^20 fixed-point scale
constexpr float INV_FXS_CAP = (1.0f / 1048576.0f) / 1.5f; // also fold /UNIT_CAP

// ---------------- workspace layout (bytes) ----------------
constexpr size_t OFF_DH   = 0;                         // 512*512 int64
constexpr size_t OFF_DV   = OFF_DH + 512*512*8;        // 512*512 int64
constexpr size_t OFF_UH   = OFF_DV + 512*512*8;        // 512*512 f32
constexpr size_t OFF_UV   = OFF_UH + 512*512*4;        // 512*512 f32
constexpr size_t OFF_PA   = OFF_UV + 512*512*4;        // node partials (2 per block)
constexpr int    NBLK_NODE  = (NN_NODES   + 255) / 256;   // 7813
constexpr int    NBLK_NET   = (NN_NETS    + 255) / 256;   // 5860
constexpr size_t OFF_PB   = OFF_PA + (size_t)NBLK_NODE * 2 * 4;
constexpr size_t OFF_SC   = OFF_PB + (size_t)NBLK_NET * 4;

typedef float v2f __attribute__((ext_vector_type(2)));
typedef float v8f __attribute__((ext_vector_type(8)));

__device__ __forceinline__ int bin_idx(float v) {
  int b = (int)floorf(v / BINW);
  return b < 0 ? 0 : (b > 511 ? 511 : b);
}

// ---------------- zero the fixed-point difference maps ----------------
__global__ void k_zero_maps(unsigned long long* D) {
  int i = blockIdx.x * 256 + threadIdx.x;
  if (i < 512 * 512 * 2) D[i] = 0ull;
}

// ---------------- per-net scatter into difference maps ----------------
__device__ __forceinline__ void scat(unsigned long long* D, int x, int y, long long v) {
  if (x < 512 && y < 512)
    atomicAdd(D + x * 512 + y, (unsigned long long)v);
}

__global__ void k_net_scatter(const float* __restrict__ pin_pos,
                              const int* __restrict__ flat_netpin,
                              unsigned long long* __restrict__ Dh,
                              unsigned long long* __restrict__ Dv) {
  int n = blockIdx.x * 256 + threadIdx.x;
  if (n >= NN_NETS) return;
  const float* py_base = pin_pos + NN_PINS;
  float xl = 1e30f, xh = -1e30f, yl = 1e30f, yh = -1e30f;
#pragma unroll
  for (int j = 0; j < 4; ++j) {
    int p = flat_netpin[4 * n + j];
    float x = pin_pos[p], y = py_base[p];
    xl = fminf(xl, x); xh = fmaxf(xh, x);
    yl = fminf(yl, y); yh = fmaxf(yh, y);
  }
  float bw = xh - xl, bh = yh - yl;
  float barea = fmaxf(bw * bh, 1e-6f);
  float dh = bw / barea, dv = bh / barea;
  int blx = bin_idx(xl), bhx = bin_idx(xh);
  int bly = bin_idx(yl), bhy = bin_idx(yh);
  long long qh = llrintf(dh * FXS);
  long long qv = llrintf(dv * FXS);
  int x1 = bhx + 1, y1 = bhy + 1;
  scat(Dh, blx, bly,  qh); scat(Dh, x1, bly, -qh);
  scat(Dh, blx, y1,  -qh); scat(Dh, x1, y1,   qh);
  scat(Dv, blx, bly,  qv); scat(Dv, x1, bly, -qv);
  scat(Dv, blx, y1,  -qv); scat(Dv, x1, y1,   qv);
}

// ---------------- fixed-point -> float (also fold /1.5 cap) ----------------
__global__ void k_convert(const unsigned long long* __restrict__ Di,
                          float* __restrict__ Uf) {
  int i = blockIdx.x * 256 + threadIdx.x;
  if (i < 512 * 512 * 2)
    Uf[i] = (float)((long long)Di[i]) * INV_FXS_CAP;
}

// ---------------- cumsum along axis 0 (x): R = LowerTri * M ----------------
// One wave per 16-wide column strip; 2 maps * 32 strips = 64 waves.
__global__ void k_cumsum_x(float* __restrict__ Uh, float* __restrict__ Uv) {
  int wid  = blockIdx.x * 8 + (threadIdx.x >> 5);   // 0..63
  int lane = threadIdx.x & 31;
  float* M = (wid & 32) ? Uv : Uh;
  int J    = wid & 31;
  int m    = lane & 15;
  bool hi  = lane >= 16;
  int col  = 16 * J + m;
  float p  = 0.0f;                                  // running column prefix
  for (int I = 0; I < 32; ++I) {
    v8f d;
#pragma unroll
    for (int r = 0; r < 8; ++r) d[r] = p;           // C: every row = p[col]
#pragma unroll
    for (int c4 = 0; c4 < 4; ++c4) {
      int kk0 = 4 * c4 + (hi ? 2 : 0);
      int kk1 = 4 * c4 + (hi ? 3 : 1);
      v2f a, b;
      a[0] = (kk0 <= m) ? 1.0f : 0.0f;              // A = LT16[:, 4c:4c+4]
      a[1] = (kk1 <= m) ? 1.0f : 0.0f;
      b[0] = M[(16 * I + kk0) * 512 + col];         // B = M[4c:4c+4, strip]
      b[1] = M[(16 * I + kk1) * 512 + col];
      d = __builtin_amdgcn_wmma_f32_16x16x4_f32(false, a, false, b,
                                                (short)0, d, false, false);
    }
#pragma unroll
    for (int r = 0; r < 8; ++r)
      M[(16 * I + r + (hi ? 8 : 0)) * 512 + col] = d[r];
    // new prefix = row 15 of D (VGPR7, lanes 16..31)
    p = __shfl(d[7], 16 + m, 32);
  }
}

// ---------------- cumsum along axis 1 (y): R = M * UpperTri ----------------
__global__ void k_cumsum_y(float* __restrict__ Uh, float* __restrict__ Uv) {
  int wid  = blockIdx.x * 8 + (threadIdx.x >> 5);
  int lane = threadIdx.x & 31;
  float* M = (wid & 32) ? Uv : Uh;
  int I    = wid & 31;
  int m    = lane & 15;
  bool hi  = lane >= 16;
  float p2[16];
#pragma unroll
  for (int r = 0; r < 16; ++r) p2[r] = 0.0f;        // running row prefixes
  for (int J = 0; J < 32; ++J) {
    v8f d;
#pragma unroll
    for (int r = 0; r < 8; ++r) d[r] = hi ? p2[r + 8] : p2[r];
#pragma unroll
    for (int c4 = 0; c4 < 4; ++c4) {
      int kk0 = 4 * c4 + (hi ? 2 : 0);
      int kk1 = 4 * c4 + (hi ? 3 : 1);
      v2f a, b;
      a[0] = M[(16 * I + m) * 512 + 16 * J + kk0];  // A = M[strip, 4c:4c+4]
      a[1] = M[(16 * I + m) * 512 + 16 * J + kk1];
      b[0] = (kk0 <= m) ? 1.0f : 0.0f;              // B = UT16[4c:4c+4, :]
      b[1] = (kk1 <= m) ? 1.0f : 0.0f;
      d = __builtin_amdgcn_wmma_f32_16x16x4_f32(false, a, false, b,
                                                (short)0, d, false, false);
    }
#pragma unroll
    for (int r = 0; r < 8; ++r)
      M[(16 * I + r + (hi ? 8 : 0)) * 512 + 16 * J + m] = d[r];
#pragma unroll
    for (int r = 0; r < 8; ++r) {                   // col 15 lives in lanes 15/31
      p2[r]     = __shfl(d[r], 15, 32);
      p2[r + 8] = __shfl(d[r], 31, 32);
    }
  }
}

// ---------------- deterministic reductions ----------------
__global__ void k_node_reduce(const float* __restrict__ nsx,
                              const float* __restrict__ nsy,
                              float* __restrict__ partials) {
  __shared__ float sm[256], sf[256];
  int i = blockIdx.x * 256 + threadIdx.x;
  float am = 0.0f, af = 0.0f;
  if (i < NN_MOVABLE) am = nsx[i] * nsy[i];
  else if (i >= NN_NODES - NN_FILLER && i < NN_NODES) af = nsx[i] * nsy[i];
  sm[threadIdx.x] = am; sf[threadIdx.x] = af; __syncthreads();
  for (int s = 128; s > 0; s >>= 1) {
    if (threadIdx.x < s) {
      sm[threadIdx.x] += sm[threadIdx.x + s];
      sf[threadIdx.x] += sf[threadIdx.x + s];
    }
    __syncthreads();
  }
  if (threadIdx.x == 0) {
    partials[2 * blockIdx.x]     = sm[0];
    partials[2 * blockIdx.x + 1] = sf[0];
  }
}

__device__ __forceinline__ float node_ratio(int i, const float* pos,
                                            const float* nsx, const float* nsy,
                                            const float* Uh, const float* Uv) {
  float cx = pos[i]             + 0.5f * nsx[i];
  float cy = pos[NN_NODES + i]  + 0.5f * nsy[i];
  int bx = bin_idx(cx), by = bin_idx(cy);
  float u = fmaxf(Uh[bx * 512 + by], Uv[bx * 512 + by]);
  return fminf(fmaxf(u, 0.5f), 2.0f);
}

__global__ void k_route_reduce(const float* __restrict__ pos,
                               const float* __restrict__ nsx,
                               const float* __restrict__ nsy,
                               const float* __restrict__ Uh,
                               const float* __restrict__ Uv,
                               float* __restrict__ partials) {
  __shared__ float sr[256];
  int i = blockIdx.x * 256 + threadIdx.x;
  float ra = 0.0f;
  if (i < NN_MOVABLE)
    ra = nsx[i] * nsy[i] * node_ratio(i, pos, nsx, nsy, Uh, Uv);
  sr[threadIdx.x] = ra; __syncthreads();
  for (int s = 128; s > 0; s >>= 1) {
    if (threadIdx.x < s) sr[threadIdx.x] += sr[threadIdx.x + s];
    __syncthreads();
  }
  if (threadIdx.x == 0) partials[blockIdx.x] = sr[0];
}

__global__ void k_scalars(const float* __restrict__ pA, int nA,
                          const float* __restrict__ pB, int nB,
                          float* __restrict__ scal) {
  __shared__ float s1[256], s2[256], s3[256];
  float a = 0.0f, b = 0.0f, c = 0.0f;
  for (int i = threadIdx.x; i < nA; i += 256) { a += pA[2 * i]; b += pA[2 * i + 1]; }
  for (int i = threadIdx.x; i < nB; i += 256) { c += pB[i]; }
  s1[threadIdx.x] = a; s2[threadIdx.x] = b; s3[threadIdx.x] = c; __syncthreads();
  for (int s = 128; s > 0; s >>= 1) {
    if (threadIdx.x < s) {
      s1[threadIdx.x] += s1[threadIdx.x + s];
      s2[threadIdx.x] += s2[threadIdx.x + s];
      s3[threadIdx.x] += s3[threadIdx.x + s];
    }
    __syncthreads();
  }
  if (threadIdx.x == 0) {
    float sum_mov = s1[0], sum_fill = s2[0], sum_route = s3[0];
    float max_total = sum_mov + sum_fill;
    float scale = fminf(1.0f, max_total / fmaxf(sum_route, 1e-6f));
    float sum_new = sum_route * scale;
    float fscale = sqrtf(fmaxf(max_total - sum_new, 0.0f) / fmaxf(sum_fill, 1e-6f));
    scal[0] = scale; scal[1] = fscale;
  }
}

// ---------------- final output ----------------
__global__ void k_output(const float* __restrict__ pos,
                         const float* __restrict__ nsx,
                         const float* __restrict__ nsy,
                         const float* __restrict__ Uh,
                         const float* __restrict__ Uv,
                         const float* __restrict__ scal,
                         float* __restrict__ out) {
  int i = blockIdx.x * 256 + threadIdx.x;
  if (i >= NN_NODES) return;
  float xi = pos[i], yi = pos[NN_NODES + i];
  float sx = nsx[i], sy = nsy[i];
  if (i < NN_MOVABLE) {
    float scale = scal[0];
    float ratio = node_ratio(i, pos, nsx, nsy, Uh, Uv);
    float area_old = sx * sy;
    float new_area = area_old * ratio * scale;
    float sr = sqrtf(new_area / fmaxf(area_old, 1e-6f));
    float nx = sx * sr, ny = sy * sr;
    out[i]                 = xi + 0.5f * (sx - nx);
    out[NN_NODES + i]      = yi + 0.5f * (sy - ny);
    out[2 * NN_NODES + i]  = nx;
    out[3 * NN_NODES + i]  = ny;
  } else if (i >= NN_NODES - NN_FILLER) {
    float fscale = scal[1];
    out[i]                 = xi;
    out[NN_NODES + i]      = yi;
    out[2 * NN_NODES + i]  = sx * fscale;
    out[3 * NN_NODES + i]  = sy * fscale;
  } else {
    out[i]                 = xi;
    out[NN_NODES + i]      = yi;
    out[2 * NN_NODES + i]  = sx;
    out[3 * NN_NODES + i]  = sy;
  }
}

// ---------------- host driver ----------------
extern "C" void kernel_launch(void* const* d_in, const int* in_sizes, int n_in,
                              void* d_out, int out_size, void* d_ws, size_t ws_size,
                              hipStream_t stream) {
  const float* pos         = (const float*)d_in[0];
  const float* pin_pos     = (const float*)d_in[1];
  const float* nsx         = (const float*)d_in[2];
  const float* nsy         = (const float*)d_in[3];
  const int*   flat_netpin = (const int*)d_in[4];
  float* out = (float*)d_out;

  char* w = (char*)d_ws;
  unsigned long long* Dh = (unsigned long long*)(w + OFF_DH);
  unsigned long long* Dv = (unsigned long long*)(w + OFF_DV);
  float* Uh   = (float*)(w + OFF_UH);
  float* Uv   = (float*)(w + OFF_UV);
  float* pA   = (float*)(w + OFF_PA);
  float* pB   = (float*)(w + OFF_PB);
  float* scal = (float*)(w + OFF_SC);

  k_zero_maps   <<<(512 * 512 * 2 + 255) / 256, 256, 0, stream>>>(Dh);
  k_net_scatter <<<NBLK_NET, 256, 0, stream>>>(pin_pos, flat_netpin, Dh, Dv);
  k_convert     <<<(512 * 512 * 2 + 255) / 256, 256, 0, stream>>>(Dh, Uh);
  k_cumsum_x    <<<8, 256, 0, stream>>>(Uh, Uv);   // 64 full waves, WMMA f32
  k_cumsum_y    <<<8, 256, 0, stream>>>(Uh, Uv);
  k_node_reduce <<<NBLK_NODE, 256, 0, stream>>>(nsx, nsy, pA);
  k_route_reduce<<<NBLK_NET, 256, 0, stream>>>(pos, nsx, nsy, Uh, Uv, pB);
  k_scalars     <<<1, 256, 0, stream>>>(pA, NBLK_NODE, pB, NBLK_NET, scal);
  k_output      <<<NBLK_NODE, 256, 0, stream>>>(pos, nsx, nsy, Uh, Uv, scal, out);
}